// MeshGCN_84576495992986
// MI455X (gfx1250) — compile-verified
//
#include <hip/hip_runtime.h>

typedef float v2f __attribute__((ext_vector_type(2)));
typedef float v8f __attribute__((ext_vector_type(8)));

// ---------------- norm precompute ----------------

__global__ void fill_f32_kernel(float* __restrict__ p, float v, int n) {
    int i = blockIdx.x * blockDim.x + threadIdx.x;
    if (i < n) p[i] = v;
}

__global__ void deg_accum_kernel(const long long* __restrict__ dst,
                                 float* __restrict__ deg, int n_edges) {
    int e = blockIdx.x * blockDim.x + threadIdx.x;
    if (e < n_edges) {
        unsafeAtomicAdd(&deg[(int)dst[e]], 1.0f);
    }
}

// deg -> dinv in place, selfnorm = dinv^2
__global__ void dinv_kernel(float* __restrict__ deg_dinv,
                            float* __restrict__ selfnorm, int n) {
    int i = blockIdx.x * blockDim.x + threadIdx.x;
    if (i < n) {
        float dg = deg_dinv[i];
        float di = (dg > 0.0f) ? rsqrtf(dg) : 0.0f;
        deg_dinv[i] = di;
        selfnorm[i] = di * di;
    }
}

// pack {src, dst, norm, 0} into one 16B record per edge (one b128 load in hot loop)
__global__ void pack_edges_kernel(const long long* __restrict__ src,
                                  const long long* __restrict__ dst,
                                  const float* __restrict__ dinv,
                                  float4* __restrict__ epack, int n_edges) {
    int e = blockIdx.x * blockDim.x + threadIdx.x;
    if (e < n_edges) {
        int s = (int)src[e];
        int d = (int)dst[e];
        float w = dinv[s] * dinv[d];
        epack[e] = make_float4(__int_as_float(s), __int_as_float(d), w, 0.0f);
    }
}

// ---------------- WMMA GEMM: Y[rows, WCOLS] = act(X)[rows, CIN] @ W[CIN, WCOLS] --------
// act = relu when RELUA (previous layer's pre-activation is the input buffer).
// One wave32 per 16x16 output tile, fp32 V_WMMA_F32_16X16X4_F32, K stepped by 4.
// A 16x4 layout : lane l holds row m = l&15, K pair at (l>>4)*2   -> v2f
// B 4x16 layout : lane l holds col n = l&15, K pair at (l>>4)*2   -> v2f
// C/D 16x16     : VGPR v, lane l -> row = v + (l>>4)*8, col = l&15
// Covers only full 16-row tiles; remainder rows handled by scalar kernel.

template <int CIN, int WCOLS, bool RELUA>
__global__ void __launch_bounds__(256)
gcn_gemm_wmma_kernel(const float* __restrict__ X,
                     const float* __restrict__ W,
                     float* __restrict__ Y,
                     int row_tiles) {
    constexpr int COL_TILES = (WCOLS + 15) / 16;
    int wave = (int)((blockIdx.x * blockDim.x + threadIdx.x) >> 5);
    if (wave >= row_tiles * COL_TILES) return;   // whole-wave exit, EXEC all-ones
    int lane = threadIdx.x & 31;

    int rt = wave / COL_TILES;                   // constexpr divisor -> shift
    int ct = wave - rt * COL_TILES;
    int row0 = rt << 4;
    int col0 = ct << 4;

    int m  = lane & 15;
    int hi = lane >> 4;                          // selects K pair {0,1} vs {2,3}
    int nb = col0 + m;                           // B column this lane supplies

    // branch-free zero-padding of B columns when WCOLS < 16 (last layer)
    int   nbc   = (WCOLS < 16) ? (nb < WCOLS ? nb : WCOLS - 1) : nb;
    float bmask = (WCOLS < 16) ? (nb < WCOLS ? 1.0f : 0.0f) : 1.0f;

    const float* xrow = X + (row0 + m) * CIN;

    v8f acc = {};
#pragma unroll
    for (int k = 0; k < CIN; k += 4) {
        int ka = k + hi * 2;
        float a0 = xrow[ka + 0];
        float a1 = xrow[ka + 1];
        if (RELUA) { a0 = fmaxf(a0, 0.0f); a1 = fmaxf(a1, 0.0f); }
        v2f a;
        a.x = a0;
        a.y = a1;
        v2f b;
        b.x = W[(ka + 0) * WCOLS + nbc] * bmask;
        b.y = W[(ka + 1) * WCOLS + nbc] * bmask;
        acc = __builtin_amdgcn_wmma_f32_16x16x4_f32(
            /*neg_a=*/false, a, /*neg_b=*/false, b,
            /*c_mod=*/(short)0, acc, /*reuse_a=*/false, /*reuse_b=*/false);
    }

    int base = (row0 + hi * 8) * WCOLS + nb;
    if (WCOLS >= 16 || nb < WCOLS) {
#pragma unroll
        for (int v = 0; v < 8; ++v)
            Y[base + v * WCOLS] = acc[v];
    }
}

// scalar fallback for rows [row_start, n_rows)  (not launched when N % 16 == 0)
__global__ void gemm_rows_naive_kernel(const float* __restrict__ X,
                                       const float* __restrict__ W,
                                       float* __restrict__ Y,
                                       int row_start, int n_rows, int cin, int cout,
                                       int relu_in) {
    int gid = blockIdx.x * blockDim.x + threadIdx.x;
    int total = (n_rows - row_start) * cout;
    if (gid >= total) return;
    int rr = gid / cout;
    int c = gid - rr * cout;
    int r = row_start + rr;
    float s = 0.0f;
    for (int k = 0; k < cin; ++k) {
        float x = X[r * cin + k];
        if (relu_in) x = fmaxf(x, 0.0f);
        s += x * W[k * cout + c];
    }
    Y[r * cout + c] = s;
}

// ---------------- agg init: agg = selfnorm*xw + bias (replaces zero-fill + finalize) ----

__global__ void init_agg64_kernel(const float4* __restrict__ xw4,
                                  const float* __restrict__ selfnorm,
                                  const float4* __restrict__ bias4,
                                  float4* __restrict__ agg4, int n_rows) {
    int gid = blockIdx.x * blockDim.x + threadIdx.x;   // n_rows * 16 float4s
    if (gid >= n_rows * 16) return;
    int node = gid >> 4;
    int ch = gid & 15;
    float sn = selfnorm[node];
    float4 x = xw4[gid];
    float4 b = bias4[ch];
    float4 r;
    r.x = sn * x.x + b.x;
    r.y = sn * x.y + b.y;
    r.z = sn * x.z + b.z;
    r.w = sn * x.w + b.w;
    agg4[gid] = r;
}

__global__ void init_agg3_kernel(const float* __restrict__ xw,
                                 const float* __restrict__ selfnorm,
                                 const float* __restrict__ bias,
                                 float* __restrict__ agg, int n_rows) {
    int gid = blockIdx.x * blockDim.x + threadIdx.x;
    if (gid >= n_rows * 3) return;
    int node = gid / 3;
    int c = gid - node * 3;
    agg[gid] = selfnorm[node] * xw[gid] + bias[c];
}

// ---------------- edge gather + scatter-add ----------------
// c_out == 64: 16 threads per edge, b128 meta + b128 gather + 4 fp32 HW atomics each.
__global__ void gcn_edge_agg64_kernel(const float4* __restrict__ epack,
                                      const float* __restrict__ xw,
                                      float* __restrict__ agg, int n_edges) {
    int gid = blockIdx.x * blockDim.x + threadIdx.x;    // E*16 = 19.2M fits int32
    int e  = gid >> 4;
    int ch = gid & 15;
    if (e >= n_edges) return;

    float4 r = epack[e];
    int s = __float_as_int(r.x);
    int d = __float_as_int(r.y);
    float w = r.z;

    float4 v = *(const float4*)(xw + (s << 6) + (ch << 2));
    float* ad = agg + (d << 6) + (ch << 2);
    unsafeAtomicAdd(ad + 0, v.x * w);
    unsafeAtomicAdd(ad + 1, v.y * w);
    unsafeAtomicAdd(ad + 2, v.z * w);
    unsafeAtomicAdd(ad + 3, v.w * w);
}

// c_out == 3: one thread per edge.
__global__ void gcn_edge_agg3_kernel(const float4* __restrict__ epack,
                                     const float* __restrict__ xw,
                                     float* __restrict__ agg, int n_edges) {
    int e = blockIdx.x * blockDim.x + threadIdx.x;
    if (e >= n_edges) return;
    float4 r = epack[e];
    int s = __float_as_int(r.x);
    int d = __float_as_int(r.y);
    float w = r.z;
    const float* xs = xw + s * 3;
    float* ad = agg + d * 3;
    unsafeAtomicAdd(ad + 0, xs[0] * w);
    unsafeAtomicAdd(ad + 1, xs[1] * w);
    unsafeAtomicAdd(ad + 2, xs[2] * w);
}

// ---------------- host ----------------

static inline int ceil_div_i(long long a, long long b) { return (int)((a + b - 1) / b); }

extern "C" void kernel_launch(void* const* d_in, const int* in_sizes, int n_in,
                              void* d_out, int out_size, void* d_ws, size_t ws_size,
                              hipStream_t stream) {
    const int CHs[7] = {8, 64, 64, 64, 64, 64, 3};
    const int Lyr = 6;

    const float* x_in = (const float*)d_in[0];
    const long long* ei = (const long long*)d_in[1];
    int N = in_sizes[0] / CHs[0];
    int E = in_sizes[1] / 2;
    const long long* src = ei;          // edge_index[0, :]
    const long long* dstv = ei + E;     // edge_index[1, :]

    const float* Wt[6];
    const float* bt[6];
    for (int i = 0; i < Lyr; ++i) {
        Wt[i] = (const float*)d_in[2 + 2 * i];
        bt[i] = (const float*)d_in[3 + 2 * i];
    }

    // workspace layout (floats; every block 16B-aligned)
    float* ws = (float*)d_ws;
    size_t off = 0;
    float* dinv  = ws + off; off += (size_t)N;        // deg, then dinv in place
    float* selfn = ws + off; off += (size_t)N;
    off = (off + 3) & ~(size_t)3;                     // align to 16B
    float4* epack = (float4*)(ws + off); off += (size_t)E * 4;
    float* xw    = ws + off; off += (size_t)N * 64;
    float* aggA  = ws + off; off += (size_t)N * 64;
    float* aggB  = ws + off; off += (size_t)N * 64;

    const int TB = 256;

    // ---- gcn_norm precompute ----
    fill_f32_kernel<<<ceil_div_i(N, TB), TB, 0, stream>>>(dinv, 1.0f, N);   // self-loops
    deg_accum_kernel<<<ceil_div_i(E, TB), TB, 0, stream>>>(dstv, dinv, E);
    dinv_kernel<<<ceil_div_i(N, TB), TB, 0, stream>>>(dinv, selfn, N);
    pack_edges_kernel<<<ceil_div_i(E, TB), TB, 0, stream>>>(src, dstv, dinv, epack, E);

    int row_tiles = N / 16;             // full tiles only
    int row_rem_start = row_tiles * 16;
    const int WPB = 8;                  // waves per block (256 threads)

    // ---- layers ----
    // Input to layer l's GEMM: pre-activation buffer from layer l-1 (relu applied on load)
    const float* cur = x_in;
    for (int l = 0; l < Lyr; ++l) {
        int cin = CHs[l];
        int cout = CHs[l + 1];

        // GEMM: xw = act(cur) @ W[l]
        if (l == 0) {
            int tiles = row_tiles * 4;
            gcn_gemm_wmma_kernel<8, 64, false><<<ceil_div_i(tiles, WPB), WPB * 32, 0, stream>>>(
                cur, Wt[l], xw, row_tiles);
        } else if (cout == 64) {
            int tiles = row_tiles * 4;
            gcn_gemm_wmma_kernel<64, 64, true><<<ceil_div_i(tiles, WPB), WPB * 32, 0, stream>>>(
                cur, Wt[l], xw, row_tiles);
        } else {
            int tiles = row_tiles;
            gcn_gemm_wmma_kernel<64, 3, true><<<ceil_div_i(tiles, WPB), WPB * 32, 0, stream>>>(
                cur, Wt[l], xw, row_tiles);
        }
        if (row_rem_start < N) {
            int total = (N - row_rem_start) * cout;
            gemm_rows_naive_kernel<<<ceil_div_i(total, TB), TB, 0, stream>>>(
                cur, Wt[l], xw, row_rem_start, N, cin, cout, l > 0 ? 1 : 0);
        }

        // agg = selfnorm*xw + bias, then scatter-add messages on top.
        // agg becomes next layer's (pre-activation) input; last layer: agg == d_out.
        float* agg = (l == Lyr - 1) ? (float*)d_out : ((l & 1) ? aggB : aggA);
        if (cout == 64) {
            init_agg64_kernel<<<ceil_div_i((long long)N * 16, TB), TB, 0, stream>>>(
                (const float4*)xw, selfn, (const float4*)bt[l], (float4*)agg, N);
            long long tot = (long long)E * 16;
            gcn_edge_agg64_kernel<<<ceil_div_i(tot, TB), TB, 0, stream>>>(
                epack, xw, agg, E);
        } else {
            init_agg3_kernel<<<ceil_div_i((long long)N * 3, TB), TB, 0, stream>>>(
                xw, selfn, bt[l], agg, N);
            gcn_edge_agg3_kernel<<<ceil_div_i(E, TB), TB, 0, stream>>>(
                epack, xw, agg, E);
        }
        cur = agg;
    }
}